// GatedAggregation_37383395345195
// MI455X (gfx1250) — compile-verified
//
#include <hip/hip_runtime.h>
#include <cstdint>
#include <cstddef>

typedef float v2f __attribute__((ext_vector_type(2)));
typedef float v8f __attribute__((ext_vector_type(8)));

#define SEGS 4096
#define DIMS 256
#define ROWS_PER_PASS 128
#define TILES_PER_PASS 8
#define LDS_STRIDE 260   // floats; 1040B row pitch: 16B aligned, conflict-free b64 A-reads
#define LDS_BYTES (ROWS_PER_PASS * LDS_STRIDE * 4)

// ---------------------------------------------------------------------------
// Kernel 1: segment boundaries via binary search on the sorted batch array.
// ---------------------------------------------------------------------------
__global__ void seg_bounds_kernel(const long long* __restrict__ batch, int N,
                                  int* __restrict__ seg_start) {
  int s = blockIdx.x * blockDim.x + threadIdx.x;
  if (s > SEGS) return;
  long long key = (long long)s;
  int lo = 0, hi = N;
  while (lo < hi) {
    int mid = (lo + hi) >> 1;
    if (batch[mid] < key) lo = mid + 1; else hi = mid;
  }
  seg_start[s] = lo;
}

// ---------------------------------------------------------------------------
// Kernel 2: one workgroup per segment (rows are contiguous since batch is
// sorted). 16 waves; wave w owns columns [16w,16w+16) of BOTH GEMMs.
// x rows are staged into LDS via GLOBAL_LOAD_ASYNC_TO_LDS_B128 (ASYNCcnt),
// dual GEMM runs on V_WMMA_F32_16X16X4_F32, and a per-column online softmax
// keeps (max, sum e, sum h*e) in registers. One pass covers 128 rows, so the
// 512KB of weights is read from L2 ~once per segment.
// ---------------------------------------------------------------------------
__global__ __launch_bounds__(512) void gated_agg_kernel(
    const float* __restrict__ x,
    const float* __restrict__ Wh, const float* __restrict__ bh,
    const float* __restrict__ Wg, const float* __restrict__ bg,
    const int* __restrict__ segs, float* __restrict__ out) {
  extern __shared__ __align__(16) float lds_x[];

  const int s  = blockIdx.x;
  const int r0 = segs[s];
  const int r1 = segs[s + 1];

  const int lane    = threadIdx.x & 31;
  const int wave    = threadIdx.x >> 5;
  const int colbase = wave * 16;
  const int n       = lane & 15;
  const int kadd    = (lane >> 4) << 1;  // K offset 0 or 2 (A/B fragment halves)
  const int half    = (lane >> 4) << 3;  // row offset 0 or 8 (C/D halves)
  const int col     = colbase + n;

  const float bhs = bh[col];
  const float bgs = bg[col];
  const float* whp = Wh + (size_t)col * DIMS + kadd;  // B[k][n] = W[col*256 + k]
  const float* wgp = Wg + (size_t)col * DIMS + kadd;
  const float* ap0 = &lds_x[n * LDS_STRIDE + kadd];
  const unsigned long long xg = (unsigned long long)(uintptr_t)x;

  const float NEG_INF = -__builtin_inff();
  float m = NEG_INF;   // running column max of g
  float dsum = 0.f;    // running sum exp(g - m)
  float num  = 0.f;    // running sum h * exp(g - m)

  for (int rt = r0; rt < r1; rt += ROWS_PER_PASS) {
    // ---- async stage of up to 128 x-rows into LDS; zero-fill the tail ----
    #pragma unroll
    for (int i = 0; i < 16; ++i) {
      int e   = threadIdx.x + i * 512;     // 0..8191 float4 slots
      int row = e >> 6;                    // / (256/4)
      int c4  = (e & 63) << 2;
      int gr  = rt + row;
      unsigned lds_off = (unsigned)(uintptr_t)(&lds_x[row * LDS_STRIDE + c4]);
      if (gr < r1) {
        unsigned goff = ((unsigned)gr * (unsigned)DIMS + (unsigned)c4) * 4u;
        asm volatile("global_load_async_to_lds_b128 %0, %1, %2"
                     :: "v"(lds_off), "v"(goff), "s"(xg) : "memory");
      } else {
        *(float4*)&lds_x[row * LDS_STRIDE + c4] = make_float4(0.f, 0.f, 0.f, 0.f);
      }
    }
    asm volatile("s_wait_asynccnt 0" ::: "memory");
    __syncthreads();

    v8f accG[TILES_PER_PASS], accH[TILES_PER_PASS];
    v8f zero = {};
    #pragma unroll
    for (int j = 0; j < TILES_PER_PASS; ++j) { accG[j] = zero; accH[j] = zero; }

    // ---- dual GEMM over K = 256 in steps of 4 ----
    #pragma unroll 2
    for (int k0 = 0; k0 < DIMS; k0 += 4) {
      v2f bgf = *(const v2f*)(wgp + k0);   // gate weights fragment (L2-resident)
      v2f bhf = *(const v2f*)(whp + k0);   // value weights fragment
      #pragma unroll
      for (int j = 0; j < TILES_PER_PASS; ++j) {
        v2f a = *(const v2f*)(ap0 + j * (16 * LDS_STRIDE) + k0);
        accG[j] = __builtin_amdgcn_wmma_f32_16x16x4_f32(
            false, a, false, bgf, (short)0, accG[j], false, false);
        accH[j] = __builtin_amdgcn_wmma_f32_16x16x4_f32(
            false, a, false, bhf, (short)0, accH[j], false, false);
      }
    }

    // ---- online softmax update, one column per lane, 8 rows per tile ----
    #pragma unroll
    for (int j = 0; j < TILES_PER_PASS; ++j) {
      int vc = r1 - (rt + j * 16);
      if (vc <= 0) break;
      if (vc > 16) vc = 16;
      float g8[8], h8[8];
      float tmax = NEG_INF;
      #pragma unroll
      for (int i = 0; i < 8; ++i) {
        bool valid = (i + half) < vc;
        float gv = accG[j][i] + bgs;
        g8[i] = valid ? gv : NEG_INF;
        h8[i] = accH[j][i] + bhs;
        tmax = fmaxf(tmax, g8[i]);
      }
      if (tmax > NEG_INF) {
        float newm = fmaxf(m, tmax);
        float sc = __expf(m - newm);       // exp(-inf)=0 on first update
        dsum *= sc; num *= sc;
        #pragma unroll
        for (int i = 0; i < 8; ++i) {
          float e = __expf(g8[i] - newm);  // 0 for masked rows
          dsum += e;
          num  += h8[i] * e;
        }
        m = newm;
      }
    }
    __syncthreads();
  }

  // ---- merge lane n with lane n+16 (same column, different row halves) ----
  float om = __shfl_xor(m,    16, 32);
  float od = __shfl_xor(dsum, 16, 32);
  float on = __shfl_xor(num,  16, 32);
  float fm = fmaxf(m, om);
  float res = 0.f;
  if (fm > NEG_INF) {
    float s0 = __expf(m - fm), s1 = __expf(om - fm);
    float Dn = dsum * s0 + od * s1;
    float Nm = num  * s0 + on * s1;
    res = (Dn != 0.f) ? (Nm / Dn) : 0.f;
  }
  if (lane < 16) out[(size_t)s * DIMS + colbase + lane] = res;
}

// ---------------------------------------------------------------------------
extern "C" void kernel_launch(void* const* d_in, const int* in_sizes, int n_in,
                              void* d_out, int out_size, void* d_ws, size_t ws_size,
                              hipStream_t stream) {
  const float*     x     = (const float*)d_in[0];
  const float*     Wh    = (const float*)d_in[1];
  const float*     bhv   = (const float*)d_in[2];
  const float*     Wg    = (const float*)d_in[3];
  const float*     bgv   = (const float*)d_in[4];
  const long long* batch = (const long long*)d_in[5];
  float* out = (float*)d_out;
  const int N = in_sizes[5];

  int* segs = (int*)d_ws;  // SEGS+1 ints of scratch

  (void)hipFuncSetAttribute((const void*)gated_agg_kernel,
                            hipFuncAttributeMaxDynamicSharedMemorySize, LDS_BYTES);

  seg_bounds_kernel<<<(SEGS + 1 + 255) / 256, 256, 0, stream>>>(batch, N, segs);
  gated_agg_kernel<<<SEGS, 512, LDS_BYTES, stream>>>(x, Wh, bhv, Wg, bgv, segs, out);
}